// Colorcal_6536940224718
// MI455X (gfx1250) — compile-verified
//
#include <hip/hip_runtime.h>

// out[b,c,h,w] = (wcam[cam[b],c] + wident[id[b],c]) * img[b,c,h,w]
//              + (bcam[cam[b],c] + bident[id[b],c])
// Pure HBM-streaming kernel: ~805 MB of traffic, ~35 us floor at 23.3 TB/s.

typedef float v4f __attribute__((ext_vector_type(4)));  // native vector: OK for NT builtins

constexpr int kB = 32;
constexpr int kC = 3;
constexpr int kH = 1024;
constexpr int kW = 1024;

constexpr int kPlaneFloats = kH * kW;            // 1,048,576 floats per (b,c) plane
constexpr int kPlaneF4     = kPlaneFloats / 4;   // 262,144 float4 per plane
constexpr int kThreads     = 256;                // 8 wave32s per block
constexpr int kF4PerThread = 4;                  // 64 B loaded + 64 B stored per thread
constexpr int kChunkF4     = kThreads * kF4PerThread;        // 1024 float4 per block
constexpr int kChunksPerPlane = kPlaneF4 / kChunkF4;         // 256 blocks per plane
constexpr int kGrid        = kB * kC * kChunksPerPlane;      // 24,576 blocks

__global__ __launch_bounds__(kThreads)
void colorcal_affine_kernel(const float* __restrict__ image,
                            const int*   __restrict__ camindex,
                            const int*   __restrict__ idindex,
                            const float* __restrict__ wcam,
                            const float* __restrict__ bcam,
                            const float* __restrict__ wident,
                            const float* __restrict__ bident,
                            float*       __restrict__ out) {
    const int plane = blockIdx.x / kChunksPerPlane;   // plane = b*3 + c, uniform
    const int chunk = blockIdx.x % kChunksPerPlane;
    const int b = plane / kC;
    const int c = plane % kC;

    // Per-plane affine params are block-uniform: force the gathered indices
    // into SGPRs so the 4 table reads become s_load and w/bias live in SGPRs.
    const int cam = __builtin_amdgcn_readfirstlane(camindex[b]);
    const int id  = __builtin_amdgcn_readfirstlane(idindex[b]);
    const float w    = wcam[cam * kC + c] + wident[id * kC + c];
    const float bias = bcam[cam * kC + c] + bident[id * kC + c];

    const size_t baseF4 = (size_t)plane * kPlaneF4
                        + (size_t)chunk * kChunkF4
                        + threadIdx.x;
    const v4f* __restrict__ src = reinterpret_cast<const v4f*>(image) + baseF4;
    v4f*       __restrict__ dst = reinterpret_cast<v4f*>(out) + baseF4;

    // Issue all loads first (4 independent global_load_b128 in flight, NT hint:
    // data streams through once, don't pollute the 192MB L2), then FMA + NT store.
    v4f v[kF4PerThread];
#pragma unroll
    for (int i = 0; i < kF4PerThread; ++i) {
        v[i] = __builtin_nontemporal_load(src + (size_t)i * kThreads);
    }
#pragma unroll
    for (int i = 0; i < kF4PerThread; ++i) {
        v4f r;
        r.x = fmaf(w, v[i].x, bias);
        r.y = fmaf(w, v[i].y, bias);
        r.z = fmaf(w, v[i].z, bias);
        r.w = fmaf(w, v[i].w, bias);
        __builtin_nontemporal_store(r, dst + (size_t)i * kThreads);
    }
}

extern "C" void kernel_launch(void* const* d_in, const int* in_sizes, int n_in,
                              void* d_out, int out_size, void* d_ws, size_t ws_size,
                              hipStream_t stream) {
    // setup_inputs() order: image, camindex, idindex, wcam, bcam, wident, bident
    const float* image    = (const float*)d_in[0];
    const int*   camindex = (const int*)  d_in[1];
    const int*   idindex  = (const int*)  d_in[2];
    const float* wcam     = (const float*)d_in[3];
    const float* bcam     = (const float*)d_in[4];
    const float* wident   = (const float*)d_in[5];
    const float* bident   = (const float*)d_in[6];
    float*       out      = (float*)d_out;

    colorcal_affine_kernel<<<kGrid, kThreads, 0, stream>>>(
        image, camindex, idindex, wcam, bcam, wident, bident, out);
}